// SAGEBlock_15281493639251
// MI455X (gfx1250) — compile-verified
//
#include <hip/hip_runtime.h>
#include <math.h>

// ---------------------------------------------------------------------------
// SAGE block for MI455X (gfx1250, wave32):
//   kernel 1: zero agg accumulator (=d_out) and per-node counts (d_ws)
//   kernel 2: edge scatter-sum via global_atomic_add_f32 (L2-resident)
//   kernel 3: fused mean + dual GEMM (fp32 WMMA 16x16x4) + GELU + LN + residual
// ---------------------------------------------------------------------------

#define FEAT 128

typedef float v2f __attribute__((ext_vector_type(2)));
typedef float v8f __attribute__((ext_vector_type(8)));

__global__ __launch_bounds__(256)
void sage_zero(float* __restrict__ agg, float* __restrict__ cnt, int n_feat_elems, int n_nodes) {
    int i = blockIdx.x * 256 + threadIdx.x;
    if (i < n_feat_elems) agg[i] = 0.0f;
    if (i < n_nodes)      cnt[i] = 0.0f;
}

// One wave (32 lanes) per edge; each lane scatters 4 features (float4 gather).
__global__ __launch_bounds__(256)
void sage_scatter(const float* __restrict__ x, const long long* __restrict__ ei,
                  int E, float* __restrict__ agg, float* __restrict__ cnt) {
    int t = blockIdx.x * 256 + threadIdx.x;
    int e = t >> 5;
    if (e >= E) return;
    int q = (t & 31) << 2;                 // feature quad 0,4,...,124
    int s = (int)ei[e];
    int d = (int)ei[(long long)E + e];
    const float4 v = *(const float4*)(x + (long long)s * FEAT + q);
    float* dst = agg + (long long)d * FEAT + q;
    atomicAdd(dst + 0, v.x);
    atomicAdd(dst + 1, v.y);
    atomicAdd(dst + 2, v.z);
    atomicAdd(dst + 3, v.w);
    if (q == 0) atomicAdd(&cnt[d], 1.0f);
}

// Block = 256 threads = 8 waves; 16 nodes (M) x 128 outputs (N) per block.
// Combined GEMM: [aggmean | x] (16x256) @ [W_l^T ; W_r^T] (256x128) via
// 64 steps of V_WMMA_F32_16X16X4_F32 per wave.
__global__ __launch_bounds__(256)
void sage_gemm_fused(const float* __restrict__ x, const float* __restrict__ aggbuf,
                     const float* __restrict__ cnt,
                     const float* __restrict__ Wl, const float* __restrict__ bl,
                     const float* __restrict__ Wr,
                     const float* __restrict__ gamma, const float* __restrict__ beta,
                     float* __restrict__ out, int N) {
    __shared__ float sA[16 * 260];          // [row][k] k=0..255 ; stride 260 -> conflict-free A gather
    __shared__ float sF[16 * 132];          // post-GELU tile
    __shared__ float sMu[16], sRs[16];

    const int t  = threadIdx.x;
    const int m0 = blockIdx.x * 16;

    // ---- stage A-panel: cols 0..127 = agg/max(cnt,1), cols 128..255 = x ----
    {
        const int row = t >> 4;             // 0..15
        const int c0  = (t & 15) * 16;      // 16 cols per thread
        const long long node = (long long)(m0 + row);
        const bool ok = (m0 + row) < N;
        const float ic = ok ? (1.0f / fmaxf(cnt[node], 1.0f)) : 0.0f;
#pragma unroll
        for (int i = 0; i < 16; ++i) {
            const int c = c0 + i;
            float v = 0.0f;
            if (ok) v = (c < FEAT) ? aggbuf[node * FEAT + c] * ic
                                   : x[node * FEAT + (c - FEAT)];
            sA[row * 260 + c] = v;
        }
    }
    __syncthreads();

    const int lane = t & 31;
    const int wave = t >> 5;                // 0..7  -> N-tile
    const int r    = lane & 15;             // A row / B col within tile
    const int koff = (lane >> 4) * 2;       // ISA 16x4 fp32 fragment K offset
    const int jcol = wave * 16 + r;         // global output column 0..127

    v8f acc = {0.f, 0.f, 0.f, 0.f, 0.f, 0.f, 0.f, 0.f};

#pragma unroll 8
    for (int kt = 0; kt < 64; ++kt) {
        const int k4 = kt * 4;
        v2f a;
        a.x = sA[r * 260 + k4 + koff];
        a.y = sA[r * 260 + k4 + koff + 1];
        // B[k][j] = W[j][k] (row-major weights, L2-resident)
        const float* __restrict__ W = (k4 < FEAT) ? Wl : Wr;
        const int kk = (k4 & (FEAT - 1)) + koff;
        v2f b;
        b.x = W[jcol * FEAT + kk];
        b.y = W[jcol * FEAT + kk + 1];
        acc = __builtin_amdgcn_wmma_f32_16x16x4_f32(
            false, a, false, b, (short)0, acc, false, false);
    }

    // ---- bias + exact GELU -> LDS tile ----
    const float bj = bl[jcol];
#pragma unroll
    for (int v = 0; v < 8; ++v) {
        const int m = (lane >> 4) * 8 + v;  // C/D layout: lanes 16..31 hold rows 8..15
        const float val = acc[v] + bj;
        const float g = 0.5f * val * (1.0f + erff(val * 0.70710678118654752f));
        sF[m * 132 + jcol] = g;
    }
    __syncthreads();

    // ---- LayerNorm stats: one thread per row ----
    if (t < 16) {
        float s = 0.0f, ss = 0.0f;
#pragma unroll 4
        for (int c = 0; c < FEAT; ++c) {
            const float v = sF[t * 132 + c];
            s += v; ss += v * v;
        }
        const float mu  = s * (1.0f / FEAT);
        const float var = ss * (1.0f / FEAT) - mu * mu;
        sMu[t] = mu;
        sRs[t] = rsqrtf(var + 1e-5f);
    }
    __syncthreads();

    // ---- normalize + gamma/beta + residual; residual x comes from staged LDS ----
    {
        const int rr = t >> 4;
        const int cc = (t & 15) * 8;
        if ((m0 + rr) < N) {
            const long long node = (long long)(m0 + rr);
            const float mu = sMu[rr], rs = sRs[rr];
#pragma unroll
            for (int i = 0; i < 8; ++i) {
                const int c = cc + i;
                const float g  = sF[rr * 132 + c];
                const float xv = sA[rr * 260 + FEAT + c];
                out[node * FEAT + c] = (g - mu) * rs * gamma[c] + beta[c] + xv;
            }
        }
    }
}

extern "C" void kernel_launch(void* const* d_in, const int* in_sizes, int n_in,
                              void* d_out, int out_size, void* d_ws, size_t ws_size,
                              hipStream_t stream) {
    const float*     x     = (const float*)d_in[0];
    const long long* ei    = (const long long*)d_in[1];   // int64 edge_index [2,E]
    const float*     Wl    = (const float*)d_in[2];
    const float*     bl    = (const float*)d_in[3];
    const float*     Wr    = (const float*)d_in[4];
    const float*     gamma = (const float*)d_in[5];
    const float*     beta  = (const float*)d_in[6];
    float*           out   = (float*)d_out;               // doubles as agg accumulator
    float*           cnt   = (float*)d_ws;                // N floats of scratch

    const int N = in_sizes[0] / FEAT;                     // 100000
    const int E = in_sizes[1] / 2;                        // 1600000

    // 1) zero accumulators
    {
        const int total = N * FEAT;
        const int blocks = (total + 255) / 256;
        sage_zero<<<blocks, 256, 0, stream>>>(out, cnt, total, N);
    }
    // 2) scatter-sum over edges (one wave per edge)
    {
        const long long work = (long long)E * 32;
        const int blocks = (int)((work + 255) / 256);
        sage_scatter<<<blocks, 256, 0, stream>>>(x, ei, E, out, cnt);
    }
    // 3) fused mean + GEMM(WMMA fp32) + GELU + LayerNorm + residual
    {
        const int blocks = (N + 15) / 16;
        sage_gemm_fused<<<blocks, 256, 0, stream>>>(x, out, cnt, Wl, bl, Wr,
                                                    gamma, beta, out, N);
    }
}